// AdaptiveTemporalEncoding_18227841204393
// MI455X (gfx1250) — compile-verified
//
#include <hip/hip_runtime.h>
#include <math.h>

// Problem constants (fixed by the reference).
#define B_   64
#define L_   4096
#define D_   128
#define D2_  64
#define NDT_ 15

typedef __attribute__((ext_vector_type(2))) float v2f;
typedef __attribute__((ext_vector_type(4))) float f4;
typedef __attribute__((ext_vector_type(8))) float v8f;

// Workspace layout (in floats):
#define OFF_CN    0                       // histogram/B : [L, 16]
#define OFF_MEANB (OFF_CN + L_ * 16)      // mean_b      : [L, 64]
#define OFF_CSUM  (OFF_MEANB + L_ * 64)   // chunk sums  : [64, 64]
#define OFF_CPRE  (OFF_CSUM + 64 * 64)    // chunk prefix: [64, 64]
#define OFF_ENC   (OFF_CPRE + 64 * 64)    // encoding    : [L, 128]
#define OFF_WT    (OFF_ENC + L_ * D_)     // padded W^T  : [64, 16]
#define WS_FLOATS (OFF_WT + 64 * 16)

// ---------------------------------------------------------------------------
// 0) Build zero-padded transposed weights: WT[n*16 + k] = W[k][n], k=15 -> 0.
//    Lets the GEMM fetch each lane's B-fragment {W[kb][n], W[kb+1][n]} as one
//    contiguous b64 load with no EXEC-masked guards.
// ---------------------------------------------------------------------------
__global__ __launch_bounds__(256) void k_prepw(const float* __restrict__ dfm,
                                               float* __restrict__ wt) {
    int idx = blockIdx.x * 256 + threadIdx.x;   // 1024 = 64 cols * 16 rows
    int k = idx & 15, n = idx >> 4;
    wt[idx] = (k < NDT_) ? dfm[k * D2_ + n] : 0.0f;
}

// ---------------------------------------------------------------------------
// 1) Per-timestep drug histogram over the batch, normalized by 1/B.
//    cn[t, d] = (#b : drugs[b,t]==d) / B   (d = 0..15, row 15 is zero pad)
// ---------------------------------------------------------------------------
__global__ __launch_bounds__(256) void k_hist(const int* __restrict__ drugs,
                                              float* __restrict__ cn) {
    int t = blockIdx.x * 256 + threadIdx.x;
    if (t >= L_) return;
    int cnt[NDT_];
#pragma unroll
    for (int d = 0; d < NDT_; ++d) cnt[d] = 0;
    for (int b = 0; b < B_; ++b) {
        int v = drugs[b * L_ + t];          // coalesced across t
#pragma unroll
        for (int d = 0; d < NDT_; ++d) cnt[d] += (v == d);
    }
    const float inv = 1.0f / (float)B_;
#pragma unroll
    for (int d = 0; d < NDT_; ++d) cn[t * 16 + d] = (float)cnt[d] * inv;
    cn[t * 16 + 15] = 0.0f;
}

// ---------------------------------------------------------------------------
// 2) mean_b[L,64] = cn[L,16] x W[16,64] via V_WMMA_F32_16X16X4_F32 (K=16 as
//    4 chained WMMAs). One wave32 per (16x16) output tile, branch-free.
//    A 16x4 f32 layout: lanes 0-15 hold {K=0,K=1}, lanes 16-31 hold {K=2,K=3}.
//    B 4x16 f32 layout: symmetric (cols across lanes); sourced from WT.
//    C/D: VGPR i -> M=i (lanes 0-15) / M=i+8 (lanes 16-31), N = lane%16.
// ---------------------------------------------------------------------------
__global__ __launch_bounds__(32) void k_gemm(const float* __restrict__ cn,
                                             const float* __restrict__ wt,
                                             float* __restrict__ meanb) {
    const int lane  = threadIdx.x;
    const int half  = lane >> 4;     // 0: lanes 0-15, 1: lanes 16-31
    const int r     = lane & 15;
    const int mtile = blockIdx.x;    // 256 tiles of 16 rows
    const int ntile = blockIdx.y;    // 4 tiles of 16 cols
    const int row   = mtile * 16 + r;   // A-row owned by this lane
    const int col   = ntile * 16 + r;   // B-col owned by this lane

    const v2f* arow = (const v2f*)(cn + row * 16);   // 8 K-pairs, b64 loads
    const v2f* bcol = (const v2f*)(wt + col * 16);   // 8 K-pairs, b64 loads

    v8f c = {};
#pragma unroll
    for (int k = 0; k < 4; ++k) {
        const int kp = 2 * k + half;    // which K-pair: kb = 4k + 2*half
        v2f a = arow[kp];
        v2f b = bcol[kp];
        c = __builtin_amdgcn_wmma_f32_16x16x4_f32(
                /*neg_a=*/false, a, /*neg_b=*/false, b,
                /*c_mod=*/(short)0, c, /*reuse_a=*/false, /*reuse_b=*/false);
    }
#pragma unroll
    for (int i = 0; i < 8; ++i) {
        const int m = mtile * 16 + half * 8 + i;
        meanb[m * D2_ + col] = c[i];
    }
}

// ---------------------------------------------------------------------------
// 3a) Per-chunk sums for the time cumsum: 64 chunks of 64 steps, 64 channels.
// ---------------------------------------------------------------------------
__global__ __launch_bounds__(256) void k_chunksum(const float* __restrict__ meanb,
                                                  float* __restrict__ csum) {
    int idx = blockIdx.x * 256 + threadIdx.x;   // 4096 = 64 chunks * 64 ch
    int c = idx >> 6, ch = idx & 63;
    float s = 0.0f;
    for (int i = 0; i < 64; ++i) s += meanb[(c * 64 + i) * D2_ + ch];
    csum[c * 64 + ch] = s;
}

// 3b) Exclusive scan of chunk sums (64 serial steps, one thread per channel).
__global__ __launch_bounds__(64) void k_scanchunks(const float* __restrict__ csum,
                                                   float* __restrict__ cpre) {
    int ch = threadIdx.x;
    float p = 0.0f;
    for (int c = 0; c < 64; ++c) {
        cpre[c * 64 + ch] = p;
        p += csum[c * 64 + ch];
    }
}

// 3c) Replay each chunk: running cumsum -> modulation -> angles -> sin/cos.
//     enc[t, 2*ch] = sin, enc[t, 2*ch+1] = cos.
__global__ __launch_bounds__(256) void k_enc(const float* __restrict__ meanb,
                                             const float* __restrict__ cpre,
                                             const float* __restrict__ bf,
                                             float* __restrict__ enc) {
    int idx = blockIdx.x * 256 + threadIdx.x;   // 4096
    int c = idx >> 6, ch = idx & 63;
    float run  = cpre[c * 64 + ch];
    float freq = bf[ch];
    for (int i = 0; i < 64; ++i) {
        int t = c * 64 + i;
        run += meanb[t * D2_ + ch];
        float mod = (t == 0) ? 0.0f : run / (float)(t + 1);
        float ang = (float)t * freq * (1.0f + mod);
        enc[t * D_ + 2 * ch]     = sinf(ang);
        enc[t * D_ + 2 * ch + 1] = cosf(ang);
    }
}

// ---------------------------------------------------------------------------
// 4) Broadcast + bolus: out[b,t,:] = enc[t,:] + bemb[:] * mask[b,t].
//    Bandwidth-bound stage: 128 MB of NT b128 stores (~5.5 us at 23.3 TB/s);
//    enc (2 MB), bemb and mask (1 MB) stay resident in L2.
// ---------------------------------------------------------------------------
__global__ __launch_bounds__(256) void k_out(const float* __restrict__ enc,
                                             const float* __restrict__ bemb,
                                             const int* __restrict__ mask,
                                             float* __restrict__ out) {
    unsigned idx = blockIdx.x * 256u + threadIdx.x;  // over B*L*(D/4) = 8M
    unsigned d4 = idx & 31u;            // which float4 of the 128-wide row
    unsigned t  = (idx >> 5) & (L_ - 1u);
    unsigned b  = idx >> 17;            // 5 (d4) + 12 (t) bits
    const f4* enc4 = (const f4*)enc;
    const f4* be4  = (const f4*)bemb;
    f4 e = enc4[t * 32u + d4];
    f4 w = be4[d4];
    float m = (mask[b * (unsigned)L_ + t] != 0) ? 1.0f : 0.0f;
    f4 o = e + w * m;
    __builtin_nontemporal_store(o, (f4*)out + (size_t)idx);
}

// ---------------------------------------------------------------------------
extern "C" void kernel_launch(void* const* d_in, const int* in_sizes, int n_in,
                              void* d_out, int out_size, void* d_ws, size_t ws_size,
                              hipStream_t stream) {
    // setup_inputs order:
    //   0: seq_len (int, 1)        1: bolus_mask (int/bool, B*L)
    //   2: dominant_drugs (int)    3: base_frequencies (float, 64)
    //   4: drug_freq_modulation (float, 15*64)
    //   5: bolus_embedding (float, 128)
    const int*   mask  = (const int*)d_in[1];
    const int*   drugs = (const int*)d_in[2];
    const float* bf    = (const float*)d_in[3];
    const float* dfm   = (const float*)d_in[4];
    const float* bemb  = (const float*)d_in[5];
    float*       out   = (float*)d_out;

    float* ws    = (float*)d_ws;
    float* cn    = ws + OFF_CN;
    float* meanb = ws + OFF_MEANB;
    float* csum  = ws + OFF_CSUM;
    float* cpre  = ws + OFF_CPRE;
    float* enc   = ws + OFF_ENC;
    float* wtp   = ws + OFF_WT;

    // 0) padded transposed weights (independent of k_hist; both tiny)
    k_prepw<<<4, 256, 0, stream>>>(dfm, wtp);
    // 1) histogram / B
    k_hist<<<L_ / 256, 256, 0, stream>>>(drugs, cn);
    // 2) WMMA GEMM: [4096,16] x [16,64] -> mean_b
    k_gemm<<<dim3(L_ / 16, D2_ / 16), 32, 0, stream>>>(cn, wtp, meanb);
    // 3) hierarchical causal cumsum + sin/cos encoding
    k_chunksum<<<L_ / 256 * 4, 256, 0, stream>>>(meanb, csum);   // 64 x 256 = 16384 threads
    k_scanchunks<<<1, 64, 0, stream>>>(csum, cpre);
    k_enc<<<L_ / 256 * 4, 256, 0, stream>>>(meanb, cpre, bf, enc);
    // 4) broadcast + bolus add (bandwidth-bound: 128 MB NT store)
    const unsigned n4 = (unsigned)B_ * L_ * (D_ / 4);            // 8,388,608
    k_out<<<n4 / 256, 256, 0, stream>>>(enc, bemb, mask, out);
}